// RecurrentProposalNetwork_35416300323735
// MI455X (gfx1250) — compile-verified
//
#include <hip/hip_runtime.h>
#include <stdint.h>

#define BATCH 16
#define NANCH 131072
#define PRE 2048
#define POST 512
#define NMS_T 0.5f
#define DW_CLIP 4.0f

typedef __attribute__((ext_vector_type(2))) float v2f;
typedef __attribute__((ext_vector_type(8))) float v8f;

__device__ __forceinline__ unsigned mono_key(float f) {
  unsigned u = __float_as_uint(f);
  return (u & 0x80000000u) ? ~u : (u | 0x80000000u);
}

// ---------------------------------------------------------------------------
// Kernel 1: per-image top-2048 via 3-level radix select (8 bits/level).
// One block per image, 1024 threads, histograms in LDS. Data stays in L2.
// ---------------------------------------------------------------------------
__global__ __launch_bounds__(1024)
void topk_select(const float* __restrict__ obj,
                 unsigned* __restrict__ kout, unsigned* __restrict__ iout) {
  const int img = blockIdx.x;
  const float* s = obj + (size_t)img * NANCH;
  __shared__ unsigned hist[256];
  __shared__ unsigned sh_prefix, sh_K, sh_cnt;
  const int tid = threadIdx.x, nt = blockDim.x;
  if (tid == 0) { sh_prefix = 0u; sh_K = PRE; sh_cnt = 0u; }

  for (int lvl = 0; lvl < 3; ++lvl) {
    const int shift = 24 - lvl * 8;
    if (tid < 256) hist[tid] = 0u;
    __syncthreads();
    const unsigned prefix = sh_prefix;
    const unsigned maskhi = (lvl == 0) ? 0u : (0xFFFFFFFFu << (shift + 8));
    for (int i = tid; i < NANCH; i += nt) {
      unsigned k = mono_key(s[i]);
      if ((k & maskhi) == prefix) atomicAdd(&hist[(k >> shift) & 255u], 1u);
    }
    __syncthreads();
    if (tid == 0) {
      unsigned K = sh_K, cum = 0u, b = 0u;
      for (int bb = 255; bb >= 0; --bb) {
        unsigned c = hist[bb];
        if (cum + c >= K) { b = (unsigned)bb; break; }
        cum += c;
      }
      sh_prefix = prefix | (b << shift);
      sh_K = K - cum;
    }
    __syncthreads();
  }
  const unsigned T = sh_prefix;  // 24-bit threshold, count(key>=T) >= PRE
  for (int i = tid; i < NANCH; i += nt) {
    unsigned k = mono_key(s[i]);
    if (k >= T) {
      unsigned pos = atomicAdd(&sh_cnt, 1u);
      if (pos < PRE) {
        kout[(size_t)img * PRE + pos] = k;
        iout[(size_t)img * PRE + pos] = (unsigned)i;
      }
    }
  }
}

// ---------------------------------------------------------------------------
// Kernel 2: per-image bitonic sort (descending) of 2048 candidates in LDS,
// then gather delta/anchor, decode 1-D boxes, clip, write to scratch.
// ---------------------------------------------------------------------------
__global__ __launch_bounds__(1024)
void sort_decode(const unsigned* __restrict__ kin, const unsigned* __restrict__ iin,
                 const float* __restrict__ delta, const float* __restrict__ anchor,
                 const int* __restrict__ imgshape, float* __restrict__ boxes_out) {
  const int img = blockIdx.x, tid = threadIdx.x;
  __shared__ unsigned skey[PRE];
  __shared__ unsigned sidx[PRE];
  skey[tid]        = kin[(size_t)img * PRE + tid];
  skey[tid + 1024] = kin[(size_t)img * PRE + tid + 1024];
  sidx[tid]        = iin[(size_t)img * PRE + tid];
  sidx[tid + 1024] = iin[(size_t)img * PRE + tid + 1024];

  for (int k = 2; k <= PRE; k <<= 1) {
    for (int j = k >> 1; j > 0; j >>= 1) {
      __syncthreads();
      for (int t = tid; t < PRE; t += 1024) {
        int l = t ^ j;
        if (l > t) {
          unsigned a = skey[t], b = skey[l];
          bool desc = ((t & k) == 0);
          if (desc ? (a < b) : (a > b)) {
            skey[t] = b; skey[l] = a;
            unsigned ia = sidx[t]; sidx[t] = sidx[l]; sidx[l] = ia;
          }
        }
      }
    }
  }
  __syncthreads();

  const float img_f = (float)imgshape[0];
  for (int t = tid; t < PRE; t += 1024) {
    unsigned idx = sidx[t];
    size_t gb = ((size_t)img * NANCH + idx) * 2;
    float d0 = delta[gb], d1 = delta[gb + 1];
    float a1 = anchor[gb], a2 = anchor[gb + 1];
    float aw = a2 - a1, ac = a1 + 0.5f * aw;
    float dw = fminf(d1, DW_CLIP);
    float pc = d0 * aw + ac;
    float pw = __expf(dw) * aw;
    float x1 = fminf(fmaxf(pc - 0.5f * pw, 0.0f), img_f);
    float x2 = fminf(fmaxf(pc + 0.5f * pw, 0.0f), img_f);
    size_t ob = ((size_t)img * PRE + t) * 2;
    boxes_out[ob] = x1; boxes_out[ob + 1] = x2;
  }
}

// ---------------------------------------------------------------------------
// Kernel 3: IoU suppression bit-mask builder.
// Grid = BATCH*16 blocks x 256 threads (8 waves). Each wave owns one 16-row
// strip and sweeps 128 column tiles. Boxes staged to LDS with async-to-LDS
// loads (ASYNCcnt). The scaled union pre-term T*(area_i + area_j) per 16x16
// tile comes off the matrix pipe via V_WMMA_F32_16X16X4_F32
// (A = [T*areas | T], B = [ones ; areas]). Per element the test is just
// min,max,sub,fma,cmp: inter > T*sum - T*inter  (<=> iou > T; inter<=0 and
// union==0 cases fall out correctly without clamps). The diagonal is cleared
// once, on the single tile where it exists (tj == rt), instead of per
// element. Row-halfword extraction from the 8 wave-uniform ballots is a
// 7-deep cndmask mux (no LDS staging, no exec churn in the hot loop).
// ---------------------------------------------------------------------------
__global__ __launch_bounds__(256)
void nms_mask_kernel(const float* __restrict__ boxes, unsigned* __restrict__ mask) {
  const int img = blockIdx.x >> 4;
  const int blk = blockIdx.x & 15;
  const int tid = threadIdx.x, lane = tid & 31, wv = tid >> 5;
  __shared__ float sbox[PRE * 2];     // 16 KB: (x1,x2) per candidate

  // ---- async copy boxes -> LDS (4 x b128 per lane) ----
  const float* gsrc = boxes + (size_t)img * PRE * 2;
#pragma unroll
  for (int it = 0; it < 4; ++it) {
    int e = (it * 256 + tid) * 4;  // float index, 16B per lane per issue
    unsigned lds_off = (unsigned)(unsigned long long)
        ((__attribute__((address_space(3))) float*)(&sbox[e]));
    unsigned long long ga = (unsigned long long)(gsrc + e);
    asm volatile("global_load_async_to_lds_b128 %0, %1, off"
                 :: "v"(lds_off), "v"(ga) : "memory");
  }
  asm volatile("s_wait_asynccnt 0" ::: "memory");
  __syncthreads();

  const int rt = blk * 8 + wv;     // row tile 0..127
  const int rbase = rt * 16;

  // A matrix 16x4 f32: lanes0-15 hold rows M=lane for K=0,1; lanes16-31 K=2,3 (zeros)
  v2f A;
  {
    int r = rbase + (lane & 15);
    float rx1 = sbox[2 * r], rx2 = sbox[2 * r + 1];
    A.x = (lane < 16) ? (NMS_T * (rx2 - rx1)) : 0.0f;  // K=0 col: T * row areas
    A.y = (lane < 16) ? NMS_T : 0.0f;                  // K=1 col: T
  }
  // per-lane row boxes for the intersection (VALU) part; C/D layout: VGPR v ->
  // M=v (lanes0-15), M=v+8 (lanes16-31)
  float rx1v[8], rx2v[8];
#pragma unroll
  for (int v = 0; v < 8; ++v) {
    int r = rbase + v + ((lane >= 16) ? 8 : 0);
    rx1v[v] = sbox[2 * r]; rx2v[v] = sbox[2 * r + 1];
  }

  const unsigned li = lane & 7;
  unsigned rowword = 0;
  for (int tj = 0; tj < 128; ++tj) {
    int c = tj * 16 + (lane & 15);
    float cx1 = sbox[2 * c], cx2 = sbox[2 * c + 1];
    float carea = cx2 - cx1;
    // B matrix 4x16 f32: row0 = ones, row1 = col areas (lanes16-31 / K=2,3 zero)
    v2f Bm;
    Bm.x = (lane < 16) ? 1.0f : 0.0f;
    Bm.y = (lane < 16) ? carea : 0.0f;
    v8f acc = {};
    acc = __builtin_amdgcn_wmma_f32_16x16x4_f32(
        /*neg_a=*/false, A, /*neg_b=*/false, Bm,
        /*c_mod=*/(short)0, acc, /*reuse_a=*/false, /*reuse_b=*/false);
    // acc[v] = T * (area_row(v) + area_col)
    unsigned bal[8];
#pragma unroll
    for (int v = 0; v < 8; ++v) {
      float inter = fminf(rx2v[v], cx2) - fmaxf(rx1v[v], cx1);
      // iou > T  <=>  inter > T*(sum - inter); inter<=0 / union==0 both give
      // "false" (keep), matching the reference's eps-guarded division.
      float rhs = __builtin_fmaf(inter, -NMS_T, acc[v]);
      bal[v] = __builtin_amdgcn_ballot_w32(inter > rhs);
    }
    // ballots are wave-uniform; mux out this lane's row halfword:
    // row l (l<8)  -> low  16 bits of bal[l]
    // row l (8..15)-> high 16 bits of bal[l-8]
    unsigned m01 = (li & 1) ? bal[1] : bal[0];
    unsigned m23 = (li & 1) ? bal[3] : bal[2];
    unsigned m45 = (li & 1) ? bal[5] : bal[4];
    unsigned m67 = (li & 1) ? bal[7] : bal[6];
    unsigned m0123 = (li & 2) ? m23 : m01;
    unsigned m4567 = (li & 2) ? m67 : m45;
    unsigned msel  = (li & 4) ? m4567 : m0123;
    unsigned half  = (lane < 8) ? (msel & 0xffffu) : (msel >> 16);
    if (tj == rt) half &= ~(1u << lane);  // clear diagonal (only tile where it exists)
    if (lane < 16) {
      if ((tj & 1) == 0) rowword = half;
      else {
        rowword |= half << 16;
        mask[((size_t)img * PRE + rbase + lane) * 64 + (tj >> 1)] = rowword;
      }
    }
  }
}

// ---------------------------------------------------------------------------
// Kernel 4: greedy scan (one wave, removed-set in 64 u32 across 32 lanes,
// wave-synchronous, zero barriers) then prefix-scan compaction matching the
// reference's order_key argsort: kept first (score order), then non-kept.
// ---------------------------------------------------------------------------
__global__ __launch_bounds__(256)
void nms_scan_compact(const unsigned* __restrict__ mask,
                      const float* __restrict__ boxes,
                      float* __restrict__ out) {
  const int img = blockIdx.x, tid = threadIdx.x, lane = tid & 31;
  __shared__ unsigned char keepA[PRE];
  __shared__ unsigned scnt[256];
  __shared__ unsigned shK;

  if (tid < 32) {
    unsigned rA = 0u, rB = 0u;  // removed words: lane, lane+32
    const unsigned* mbase = mask + (size_t)img * PRE * 64;
    for (int i = 0; i < PRE; ++i) {
      unsigned wvn = (unsigned)i >> 5;
      unsigned wa = __shfl(rA, (int)(wvn & 31u));
      unsigned wb = __shfl(rB, (int)(wvn & 31u));
      unsigned wordv = (wvn < 32u) ? wa : wb;
      bool keep_i = ((wordv >> (i & 31)) & 1u) == 0u;
      if (keep_i) {
        const unsigned* row = mbase + (size_t)i * 64;
        rA |= row[lane];
        rB |= row[lane + 32];
      }
      if (lane == 0) keepA[i] = keep_i ? 1 : 0;
    }
  }
  __syncthreads();

  unsigned cnt = 0;
#pragma unroll
  for (int j = 0; j < 8; ++j) cnt += keepA[tid * 8 + j];
  scnt[tid] = cnt;
  __syncthreads();
  for (int off = 1; off < 256; off <<= 1) {
    unsigned v = scnt[tid];
    unsigned add = (tid >= off) ? scnt[tid - off] : 0u;
    __syncthreads();
    scnt[tid] = v + add;
    __syncthreads();
  }
  unsigned incl = scnt[tid];
  unsigned excl = incl - cnt;
  if (tid == 255) shK = incl;
  __syncthreads();
  const unsigned K = shK;

  float* prop = out;                               // [B,512,2]
  float* sel  = out + BATCH * POST * 2;            // [B,512]
  float* val  = out + BATCH * POST * 2 + BATCH * POST;  // [B,512]

  unsigned kr = excl;
  for (int j = 0; j < 8; ++j) {
    int i = tid * 8 + j;
    bool kp = keepA[i] != 0;
    unsigned slot = kp ? kr : (K + ((unsigned)i - kr));
    if (kp) kr++;
    if (slot < POST) {
      size_t ob = (size_t)img * POST + slot;
      float x1 = kp ? boxes[((size_t)img * PRE + i) * 2]     : 0.0f;
      float x2 = kp ? boxes[((size_t)img * PRE + i) * 2 + 1] : 0.0f;
      prop[ob * 2]     = x1;
      prop[ob * 2 + 1] = x2;
      sel[ob] = (float)i;
      val[ob] = kp ? 1.0f : 0.0f;
    }
  }
}

// ---------------------------------------------------------------------------
extern "C" void kernel_launch(void* const* d_in, const int* in_sizes, int n_in,
                              void* d_out, int out_size, void* d_ws, size_t ws_size,
                              hipStream_t stream) {
  const float* obj    = (const float*)d_in[0];
  const float* delta  = (const float*)d_in[1];
  const float* anchor = (const float*)d_in[2];
  const int*   imgsh  = (const int*)d_in[3];
  float* out = (float*)d_out;

  char* ws = (char*)d_ws;
  unsigned* ws_key  = (unsigned*)(ws);                 // B*2048 u32
  unsigned* ws_idx  = (unsigned*)(ws + 131072);        // B*2048 u32
  float*    ws_box  = (float*)   (ws + 262144);        // B*2048*2 f32
  unsigned* ws_mask = (unsigned*)(ws + 524288);        // B*2048*64 u32 (8 MB)

  topk_select<<<BATCH, 1024, 0, stream>>>(obj, ws_key, ws_idx);
  sort_decode<<<BATCH, 1024, 0, stream>>>(ws_key, ws_idx, delta, anchor, imgsh, ws_box);
  nms_mask_kernel<<<BATCH * 16, 256, 0, stream>>>(ws_box, ws_mask);
  nms_scan_compact<<<BATCH, 256, 0, stream>>>(ws_mask, ws_box, out);
}